// GraphAttentionNetwork_31963146617554
// MI455X (gfx1250) — compile-verified
//
#include <hip/hip_runtime.h>
#include <cmath>

#define N_NODES 100000
#define N_EDGES 1600000
#define E_TOT   (N_EDGES + N_NODES)   // self loops appended
#define NEG_SLOPE 0.2f
#define LN_EPS 1e-5f

typedef __attribute__((ext_vector_type(2))) float v2f;
typedef __attribute__((ext_vector_type(8))) float v8f;

// ---------------------------------------------------------------------------
// helpers
// ---------------------------------------------------------------------------
__device__ __forceinline__ void atomicMaxF(float* addr, float v) {
  // sign-magnitude ordering trick: correct float max for mixed signs
  if (v >= 0.f) atomicMax((int*)addr, __float_as_int(v));
  else          atomicMin((unsigned int*)addr, __float_as_uint(v));
}

__device__ __forceinline__ void edge_sd(const long long* __restrict__ ei, int e,
                                        int& s, int& d) {
  if (e < N_EDGES) { s = (int)ei[e]; d = (int)ei[N_EDGES + e]; }
  else             { s = e - N_EDGES; d = s; }           // self loop
}

__global__ void fill_kernel(float* __restrict__ p, float v, int n) {
  int t = blockIdx.x * blockDim.x + threadIdx.x;
  if (t < n) p[t] = v;
}

// ---------------------------------------------------------------------------
// GEMM via V_WMMA_F32_16X16X4_F32: C[M x 64] = A[M x K] * B[K x 64]
// one wave -> 16-row strip x 64 cols; B staged in LDS
// ---------------------------------------------------------------------------
template<int K>
__global__ void gemm_wmma_kernel(const float* __restrict__ A,
                                 const float* __restrict__ B,
                                 float* __restrict__ C, int M) {
  __shared__ float ldsB[K * 64];
  for (int i = threadIdx.x; i < K * 64; i += blockDim.x) ldsB[i] = B[i];
  __syncthreads();

  const int wave = threadIdx.x >> 5;
  const int lane = threadIdx.x & 31;
  const int rt   = blockIdx.x * 8 + wave;        // 16-row tile index
  if (rt * 16 >= M) return;                      // wave-uniform guard (EXEC stays full)

  const int row_base = rt * 16;
  const int mrow  = lane & 15;                   // A: M = lane%16
  const int khalf = (lane >> 4) << 1;            // A/B: K pair = 0 or 2
  const int ncol  = lane & 15;                   // B/C: N = lane%16

  v8f acc0 = {}, acc1 = {}, acc2 = {}, acc3 = {};
  const float* arow = A + (size_t)(row_base + mrow) * K + khalf;

#pragma unroll
  for (int k = 0; k < K; k += 4) {
    v2f a; a.x = arow[k]; a.y = arow[k + 1];
    const float* bp = &ldsB[(k + khalf) * 64];
    v2f b0, b1, b2, b3;
    b0.x = bp[ncol     ]; b0.y = bp[64 + ncol     ];
    b1.x = bp[ncol + 16]; b1.y = bp[64 + ncol + 16];
    b2.x = bp[ncol + 32]; b2.y = bp[64 + ncol + 32];
    b3.x = bp[ncol + 48]; b3.y = bp[64 + ncol + 48];
    acc0 = __builtin_amdgcn_wmma_f32_16x16x4_f32(false, a, false, b0, (short)0, acc0, false, false);
    acc1 = __builtin_amdgcn_wmma_f32_16x16x4_f32(false, a, false, b1, (short)0, acc1, false, false);
    acc2 = __builtin_amdgcn_wmma_f32_16x16x4_f32(false, a, false, b2, (short)0, acc2, false, false);
    acc3 = __builtin_amdgcn_wmma_f32_16x16x4_f32(false, a, false, b3, (short)0, acc3, false, false);
  }

  const int mofs = (lane >> 4) << 3;             // C: M = v + 8*(lane/16)
  float* crow = C + (size_t)(row_base + mofs) * 64 + ncol;
#pragma unroll
  for (int v = 0; v < 8; ++v) {
    crow[(size_t)v * 64 +  0] = acc0[v];
    crow[(size_t)v * 64 + 16] = acc1[v];
    crow[(size_t)v * 64 + 32] = acc2[v];
    crow[(size_t)v * 64 + 48] = acc3[v];
  }
}

// ---------------------------------------------------------------------------
// per-node attention scores, layer 1: thread per (node,head), C=16
// ---------------------------------------------------------------------------
__global__ void scores1_kernel(const float* __restrict__ h1,
                               const float* __restrict__ att_src,
                               const float* __restrict__ att_dst,
                               float* __restrict__ asrc, float* __restrict__ adst) {
  int t = blockIdx.x * blockDim.x + threadIdx.x;          // node*4 + head
  if (t >= N_NODES * 4) return;
  int hd = t & 3, n = t >> 2;
  const float* hp = h1 + (size_t)n * 64 + hd * 16;
  const float* as = att_src + hd * 16;
  const float* ad = att_dst + hd * 16;
  float s = 0.f, d = 0.f;
#pragma unroll
  for (int c = 0; c < 16; ++c) { s += hp[c] * as[c]; d += hp[c] * ad[c]; }
  asrc[t] = s; adst[t] = d;
}

// layer 2: thread per node, C=64, 1 head
__global__ void scores2_kernel(const float* __restrict__ h2,
                               const float* __restrict__ att_src,
                               const float* __restrict__ att_dst,
                               float* __restrict__ asrc, float* __restrict__ adst) {
  int n = blockIdx.x * blockDim.x + threadIdx.x;
  if (n >= N_NODES) return;
  const float* hp = h2 + (size_t)n * 64;
  float s = 0.f, d = 0.f;
#pragma unroll 8
  for (int c = 0; c < 64; ++c) { s += hp[c] * att_src[c]; d += hp[c] * att_dst[c]; }
  asrc[n] = s; adst[n] = d;
}

// ---------------------------------------------------------------------------
// edge phase: leaky_relu score -> segment max -> exp / segment sum -> scatter
// ---------------------------------------------------------------------------
template<int NH>
__global__ void edge_max_kernel(const long long* __restrict__ ei,
                                const float* __restrict__ asrc,
                                const float* __restrict__ adst,
                                float* __restrict__ eraw,
                                float* __restrict__ mbuf) {
  int t = blockIdx.x * blockDim.x + threadIdx.x;
  if (t >= E_TOT * NH) return;
  int h = (NH == 1) ? 0 : (t & 3);
  int e = (NH == 1) ? t : (t >> 2);
  int s, d; edge_sd(ei, e, s, d);
  float v = asrc[s * NH + h] + adst[d * NH + h];
  v = (v > 0.f) ? v : NEG_SLOPE * v;
  eraw[t] = v;
  atomicMaxF(&mbuf[d * NH + h], v);
}

template<int NH>
__global__ void edge_expsum_kernel(const long long* __restrict__ ei,
                                   const float* __restrict__ mbuf,
                                   float* __restrict__ eraw,
                                   float* __restrict__ sbuf) {
  int t = blockIdx.x * blockDim.x + threadIdx.x;
  if (t >= E_TOT * NH) return;
  int h = (NH == 1) ? 0 : (t & 3);
  int e = (NH == 1) ? t : (t >> 2);
  int s, d; edge_sd(ei, e, s, d);
  float v = expf(eraw[t] - mbuf[d * NH + h]);
  eraw[t] = v;
  atomicAdd(&sbuf[d * NH + h], v);
}

// layer1 scatter: thread per (edge, head); 16 channels each
__global__ void scatter1_kernel(const long long* __restrict__ ei,
                                const float* __restrict__ eexp,
                                const float* __restrict__ sbuf,
                                const float* __restrict__ h1,
                                float* __restrict__ acc) {
  int t = blockIdx.x * blockDim.x + threadIdx.x;
  if (t >= E_TOT * 4) return;
  int h = t & 3, e = t >> 2;
  int s, d; edge_sd(ei, e, s, d);
  float alpha = eexp[t] / (sbuf[d * 4 + h] + 1e-16f);
  const float* hp = h1 + (size_t)s * 64 + h * 16;
  float* op = acc + (size_t)d * 64 + h * 16;
#pragma unroll
  for (int c = 0; c < 16; ++c) atomicAdd(&op[c], hp[c] * alpha);
}

// layer2 scatter: 4 threads per edge; 16 of 64 channels each
__global__ void scatter2_kernel(const long long* __restrict__ ei,
                                const float* __restrict__ eexp,
                                const float* __restrict__ sbuf,
                                const float* __restrict__ h2,
                                float* __restrict__ acc) {
  int t = blockIdx.x * blockDim.x + threadIdx.x;
  if (t >= E_TOT * 4) return;
  int p = t & 3, e = t >> 2;
  int s, d; edge_sd(ei, e, s, d);
  float alpha = eexp[e] / (sbuf[d] + 1e-16f);
  const float* hp = h2 + (size_t)s * 64 + p * 16;
  float* op = acc + (size_t)d * 64 + p * 16;
#pragma unroll
  for (int c = 0; c < 16; ++c) atomicAdd(&op[c], hp[c] * alpha);
}

// ---------------------------------------------------------------------------
// bias + LayerNorm(64) (+optional ReLU): one wave32 per node, shfl reductions
// ---------------------------------------------------------------------------
__global__ void ln_kernel(const float* __restrict__ acc,
                          const float* __restrict__ bias,
                          const float* __restrict__ g,
                          const float* __restrict__ bln,
                          float* __restrict__ out, int do_relu) {
  int wave = threadIdx.x >> 5, lane = threadIdx.x & 31;
  int n = blockIdx.x * (blockDim.x >> 5) + wave;
  if (n >= N_NODES) return;
  const float* row = acc + (size_t)n * 64;
  float v0 = row[lane]      + bias[lane];
  float v1 = row[lane + 32] + bias[lane + 32];
  float s = v0 + v1, ss = v0 * v0 + v1 * v1;
#pragma unroll
  for (int m = 16; m >= 1; m >>= 1) {
    s  += __shfl_xor(s,  m, 32);
    ss += __shfl_xor(ss, m, 32);
  }
  float mu  = s * (1.f / 64.f);
  float var = ss * (1.f / 64.f) - mu * mu;
  float inv = rsqrtf(var + LN_EPS);
  float y0 = (v0 - mu) * inv * g[lane]      + bln[lane];
  float y1 = (v1 - mu) * inv * g[lane + 32] + bln[lane + 32];
  if (do_relu) { y0 = fmaxf(y0, 0.f); y1 = fmaxf(y1, 0.f); }
  out[(size_t)n * 64 + lane]      = y0;
  out[(size_t)n * 64 + lane + 32] = y1;
}

// ---------------------------------------------------------------------------
// launch
// ---------------------------------------------------------------------------
extern "C" void kernel_launch(void* const* d_in, const int* in_sizes, int n_in,
                              void* d_out, int out_size, void* d_ws, size_t ws_size,
                              hipStream_t stream) {
  const float*     x        = (const float*)d_in[0];
  const long long* ei       = (const long long*)d_in[1];
  const float*     W1       = (const float*)d_in[2];
  const float*     att_src1 = (const float*)d_in[3];
  const float*     att_dst1 = (const float*)d_in[4];
  const float*     b1       = (const float*)d_in[5];
  const float*     ln1g     = (const float*)d_in[6];
  const float*     ln1b     = (const float*)d_in[7];
  const float*     W2       = (const float*)d_in[8];
  const float*     att_src2 = (const float*)d_in[9];
  const float*     att_dst2 = (const float*)d_in[10];
  const float*     b2       = (const float*)d_in[11];
  const float*     ln2g     = (const float*)d_in[12];
  const float*     ln2b     = (const float*)d_in[13];

  float* ws    = (float*)d_ws;
  float* h1    = ws;                 // 6,400,000 f32
  float* asrc1 = ws +  6400000;      //   400,000
  float* adst1 = ws +  6800000;      //   400,000
  float* m1    = ws +  7200000;      //   400,000
  float* s1    = ws +  7600000;      //   400,000
  float* e1    = ws +  8000000;      // 6,800,000 (E_TOT*4)
  float* acc1  = ws + 14800000;      // 6,400,000   (peak ws ~84.8 MB)
  // layer-2 reuse (lifetimes disjoint):
  float* hln   = e1;                 // 6,400,000  (h1 strip dead after scatter1? no: e1 dead)
  float* h2    = h1;                 // h1 dead after scatter1
  float* asrc2 = asrc1; float* adst2 = adst1;
  float* m2    = m1;    float* s2    = s1;
  float* e2    = acc1;               // acc1 dead after ln_kernel(1)
  float* acc2  = (float*)d_out;      // accumulate layer-2 output in-place

  const int TB = 256;
  const int nTiles   = N_NODES / 16;                 // 6250 (exact)
  const int gemmGrid = (nTiles + 7) / 8;
  const int e4Grid   = (E_TOT * 4 + TB - 1) / TB;
  const int e1Grid   = (E_TOT + TB - 1) / TB;
  const int lnGrid   = (N_NODES + 7) / 8;

  // ---- layer 1 ----
  fill_kernel<<<(400000 + TB - 1) / TB, TB, 0, stream>>>(m1, -INFINITY, 400000);
  fill_kernel<<<(400000 + TB - 1) / TB, TB, 0, stream>>>(s1, 0.f, 400000);
  fill_kernel<<<(6400000 + TB - 1) / TB, TB, 0, stream>>>(acc1, 0.f, 6400000);

  gemm_wmma_kernel<128><<<gemmGrid, 256, 0, stream>>>(x, W1, h1, N_NODES);
  scores1_kernel<<<(N_NODES * 4 + TB - 1) / TB, TB, 0, stream>>>(h1, att_src1, att_dst1, asrc1, adst1);
  edge_max_kernel<4><<<e4Grid, TB, 0, stream>>>(ei, asrc1, adst1, e1, m1);
  edge_expsum_kernel<4><<<e4Grid, TB, 0, stream>>>(ei, m1, e1, s1);
  scatter1_kernel<<<e4Grid, TB, 0, stream>>>(ei, e1, s1, h1, acc1);
  ln_kernel<<<lnGrid, 256, 0, stream>>>(acc1, b1, ln1g, ln1b, hln, 1);

  // ---- layer 2 ----
  gemm_wmma_kernel<64><<<gemmGrid, 256, 0, stream>>>(hln, W2, h2, N_NODES);
  scores2_kernel<<<(N_NODES + TB - 1) / TB, TB, 0, stream>>>(h2, att_src2, att_dst2, asrc2, adst2);
  fill_kernel<<<(N_NODES + TB - 1) / TB, TB, 0, stream>>>(m2, -INFINITY, N_NODES);
  fill_kernel<<<(N_NODES + TB - 1) / TB, TB, 0, stream>>>(s2, 0.f, N_NODES);
  fill_kernel<<<(6400000 + TB - 1) / TB, TB, 0, stream>>>(acc2, 0.f, 6400000);
  edge_max_kernel<1><<<e1Grid, TB, 0, stream>>>(ei, asrc2, adst2, e2, m2);
  edge_expsum_kernel<1><<<e1Grid, TB, 0, stream>>>(ei, m2, e2, s2);
  scatter2_kernel<<<e4Grid, TB, 0, stream>>>(ei, e2, s2, h2, acc2);
  ln_kernel<<<lnGrid, 256, 0, stream>>>(acc2, b2, ln2g, ln2b, acc2, 0);
}